// Grapher_42082089566466
// MI455X (gfx1250) — compile-verified
//
// MI455X (gfx1250) Grapher layer: fc1 -> KNN(EdgeConv) -> BN -> max-k -> fc2 + residual
//
// Design notes: EdgeConv linearized via P = h(Wa-Wb), Q = h*Wb (174 GFLOP GEMM ->
// 19 GFLOP + gathers); all GEMMs + the N^2 distance matrix use
// V_WMMA_F32_16X16X4_F32 (native fp32 matrix path). KNN is fused tiled-distance
// + in-register top-9 insertion (never materializes the 1 GB N x N matrix).
// LDS tiles use even padded strides so every WMMA fragment is a single aligned
// ds_load_b64 into an even VGPR pair (no repacking moves). Deterministic
// two-pass reductions (no float atomics). Workspace ~206 MB.

#include <hip/hip_runtime.h>
#include <hip/hip_bf16.h>
#include <math.h>

#define NPTS 16384
#define CH   384
#define CH2  768
#define KNN  9
#define BN_EPS 1e-5f
#define FLT_BIG 3.4028235e38f

typedef __attribute__((ext_vector_type(2))) float v2f;
typedef __attribute__((ext_vector_type(8))) float v8f;

__device__ __forceinline__ v8f wmma_f32_16x16x4(v2f a, v2f b, v8f c) {
  // 8 args: (neg_a, A, neg_b, B, c_mod, C, reuse_a, reuse_b)
  return __builtin_amdgcn_wmma_f32_16x16x4_f32(false, a, false, b, (short)0, c,
                                               false, false);
}

__device__ __forceinline__ v2f lds_pair(const float* p) {
  return *(const v2f*)p;  // 8B-aligned by construction -> ds_load_b64
}

// ---------------------------------------------------------------- prep kernels
__global__ void transpose_kernel(const float* __restrict__ src,
                                 float* __restrict__ dst, int rows, int cols) {
  // dst[c][r] = src[r][c];  dst is [cols][rows]
  int i = blockIdx.x * blockDim.x + threadIdx.x;
  int total = rows * cols;
  if (i < total) {
    int r = i / cols, c = i - r * cols;
    dst[c * rows + r] = src[i];
  }
}

__global__ void split_wconv_kernel(const float* __restrict__ wconv,
                                   float* __restrict__ wb1,
                                   float* __restrict__ wb2) {
  // wconv: [CH2 out][CH2 in].  wb1[c][o] = wconv[o][c] - wconv[o][CH+c]
  //                            wb2[c][o] = wconv[o][CH+c]     (c < CH)
  int i = blockIdx.x * blockDim.x + threadIdx.x;
  if (i < CH * CH2) {
    int c = i / CH2, o = i - c * CH2;
    float wa = wconv[o * CH2 + c];
    float wb = wconv[o * CH2 + CH + c];
    wb1[c * CH2 + o] = wa - wb;
    wb2[c * CH2 + o] = wb;
  }
}

// ------------------------------------------------------------ generic WMMA GEMM
// out[M,Nc] = op(A[M,Kc] @ B[Kc,Nc] + bias) (+ resid); op = relu if do_relu.
// As: [m][k] stride 66;  Bs: [n][k] stride 66 (transposed at staging) so that
// each fragment (k, k+1 at even k) is one aligned ds_load_b64, bank-conflict-free.
#define GEMM_BK 64
#define GEMM_ST 66
__global__ __launch_bounds__(128) void gemm_wmma_kernel(
    const float* __restrict__ A, const float* __restrict__ B,
    const float* __restrict__ bias, const float* __restrict__ resid,
    float* __restrict__ out, int M, int Kc, int Nc, int do_relu) {
  __shared__ __align__(16) float As[64 * GEMM_ST];
  __shared__ __align__(16) float Bs[64 * GEMM_ST];
  const int tid = threadIdx.x;
  const int lane = tid & 31, wv = tid >> 5;
  const int l16 = lane & 15, hi = lane >> 4, h2 = hi << 1;
  const int m0 = blockIdx.x * 64;
  const int n0 = blockIdx.y * 64;

  const v8f z8 = {0.f, 0.f, 0.f, 0.f, 0.f, 0.f, 0.f, 0.f};
  v8f acc[4];
#pragma unroll
  for (int nt = 0; nt < 4; ++nt) acc[nt] = z8;

  for (int k0 = 0; k0 < Kc; k0 += GEMM_BK) {
    for (int i = tid; i < 64 * GEMM_BK; i += 128) {
      int r = i >> 6, c = i & 63;
      As[r * GEMM_ST + c] = A[(size_t)(m0 + r) * Kc + k0 + c];
      // B row r is k-index, col c is n-index: store transposed -> Bs[n][k]
      Bs[c * GEMM_ST + r] = B[(size_t)(k0 + r) * Nc + n0 + c];
    }
    __syncthreads();
    if (k0 + GEMM_BK < Kc)  // prefetch next B tile (global_prefetch_b8)
      __builtin_prefetch(&B[(size_t)(k0 + GEMM_BK + (tid >> 6)) * Nc + n0 + (tid & 63)], 0, 1);
#pragma unroll
    for (int kk = 0; kk < GEMM_BK / 4; ++kk) {
      const int kb = kk * 4 + h2;  // even -> 8B aligned pairs
      const v2f a = lds_pair(&As[(wv * 16 + l16) * GEMM_ST + kb]);
#pragma unroll
      for (int nt = 0; nt < 4; ++nt) {
        const v2f b = lds_pair(&Bs[(nt * 16 + l16) * GEMM_ST + kb]);
        acc[nt] = wmma_f32_16x16x4(a, b, acc[nt]);
      }
    }
    __syncthreads();
  }

  const int mbase = m0 + wv * 16 + (hi << 3);
#pragma unroll
  for (int nt = 0; nt < 4; ++nt) {
    const int col = n0 + nt * 16 + l16;
    const float bv = bias ? bias[col] : 0.f;
#pragma unroll
    for (int v = 0; v < 8; ++v) {
      const int row = mbase + v;
      float val = acc[nt][v] + bv;
      if (do_relu) val = fmaxf(val, 0.f);
      if (resid) val += resid[(size_t)row * Nc + col];
      out[(size_t)row * Nc + col] = val;
    }
  }
}

// ---------------------------------------------------- column norms (2-pass det.)
__global__ __launch_bounds__(128) void colsumsq_kernel(
    const float* __restrict__ h, float* __restrict__ colpart) {
  int t = threadIdx.x;
  int r0 = blockIdx.x * 64;
  float s0 = 0.f, s1 = 0.f, s2 = 0.f;
  for (int r = 0; r < 64; ++r) {
    const float* row = h + (size_t)(r0 + r) * CH;
    float a = row[t], b = row[t + 128], c = row[t + 256];
    s0 += a * a; s1 += b * b; s2 += c * c;
  }
  colpart[blockIdx.x * CH + t]       = s0;
  colpart[blockIdx.x * CH + t + 128] = s1;
  colpart[blockIdx.x * CH + t + 256] = s2;
}

__global__ void colreduce_kernel(const float* __restrict__ colpart,
                                 float* __restrict__ invn) {
  int c = blockIdx.x * blockDim.x + threadIdx.x;
  if (c < CH) {
    float s = 0.f;
    for (int b = 0; b < NPTS / 64; ++b) s += colpart[b * CH + c];
    invn[c] = 1.f / fmaxf(sqrtf(s), 1e-12f);
  }
}

__global__ __launch_bounds__(128) void rownorm_kernel(
    const float* __restrict__ h, const float* __restrict__ invn,
    float* __restrict__ hn, float* __restrict__ sqrow) {
  __shared__ float red[128];
  int t = threadIdx.x;
  size_t base = (size_t)blockIdx.x * CH;
  float part = 0.f;
#pragma unroll
  for (int i = 0; i < 3; ++i) {
    int c = t + i * 128;
    float v = h[base + c] * invn[c];
    hn[base + c] = v;
    part += v * v;
  }
  red[t] = part;
  __syncthreads();
  for (int s = 64; s > 0; s >>= 1) {
    if (t < s) red[t] += red[t + s];
    __syncthreads();
  }
  if (t == 0) sqrow[blockIdx.x] = red[0];
}

// ------------------------------------------------- fused tiled KNN (WMMA + top9)
#define KNN_CK 96
#define KNN_ST 98  // even padded stride: aligned b64 frags, conflict-free (34*r mod 64 distinct)
__global__ __launch_bounds__(128) void knn_kernel(
    const float* __restrict__ hn, const float* __restrict__ sqrow,
    int* __restrict__ knn_idx) {
  __shared__ __align__(16) float As[64 * KNN_ST];
  __shared__ __align__(16) float Bs[64 * KNN_ST];
  __shared__ float distS[4 * 16 * 16];
  __shared__ float mrgD[4 * 16 * 18];
  __shared__ int   mrgI[4 * 16 * 18];

  const int tid = threadIdx.x;
  const int lane = tid & 31, wv = tid >> 5;
  const int l16 = lane & 15, hi = lane >> 4, h2 = hi << 1;
  const int i0 = blockIdx.x * 64;

  float sqI[8];
#pragma unroll
  for (int v = 0; v < 8; ++v) sqI[v] = sqrow[i0 + wv * 16 + (hi << 3) + v];

  // per-lane top-9 (lane l16 owns row l16 of its wave's stripe; hi selects col half)
  float ld[KNN]; int li[KNN];
#pragma unroll
  for (int s = 0; s < KNN; ++s) { ld[s] = FLT_BIG; li[s] = 0x7fffffff; }

  for (int tile = 0; tile < NPTS / 64; ++tile) {
    const int j0 = tile * 64;
    const v8f z8 = {0.f, 0.f, 0.f, 0.f, 0.f, 0.f, 0.f, 0.f};
    v8f acc[4];
#pragma unroll
    for (int nt = 0; nt < 4; ++nt) acc[nt] = z8;

    for (int c0 = 0; c0 < CH; c0 += KNN_CK) {
      __syncthreads();
      for (int i = tid; i < 64 * KNN_CK; i += 128) {
        int r = i / KNN_CK, c = i - r * KNN_CK;
        As[r * KNN_ST + c] = hn[(size_t)(i0 + r) * CH + c0 + c];
        Bs[r * KNN_ST + c] = hn[(size_t)(j0 + r) * CH + c0 + c];
      }
      __syncthreads();
#pragma unroll
      for (int kk = 0; kk < KNN_CK / 4; ++kk) {
        const int kb = kk * 4 + h2;  // even -> 8B aligned pairs
        const v2f a = lds_pair(&As[(wv * 16 + l16) * KNN_ST + kb]);
#pragma unroll
        for (int nt = 0; nt < 4; ++nt) {
          const v2f b = lds_pair(&Bs[(nt * 16 + l16) * KNN_ST + kb]);
          acc[nt] = wmma_f32_16x16x4(a, b, acc[nt]);
        }
      }
    }

    // dist = sq_i - 2*dot + sq_j ; transpose via per-wave LDS tile, then top-9 insert
#pragma unroll
    for (int nt = 0; nt < 4; ++nt) {
      const float sqJ = sqrow[j0 + nt * 16 + l16];
#pragma unroll
      for (int v = 0; v < 8; ++v) {
        const int Ml = (hi << 3) + v;
        distS[wv * 256 + Ml * 16 + l16] = fmaf(-2.f, acc[nt][v], sqI[v] + sqJ);
      }
      // LDS ops from one wave are processed in-order: stores above are visible
#pragma unroll
      for (int c = 0; c < 8; ++c) {
        const int cc = hi * 8 + c;
        const float d = distS[wv * 256 + l16 * 16 + cc];
        const int j = j0 + nt * 16 + cc;
        if (d < ld[KNN - 1]) {
          ld[KNN - 1] = d; li[KNN - 1] = j;
#pragma unroll
          for (int s = KNN - 1; s >= 1; --s) {
            if (ld[s] < ld[s - 1]) {
              float td = ld[s]; ld[s] = ld[s - 1]; ld[s - 1] = td;
              int ti = li[s]; li[s] = li[s - 1]; li[s - 1] = ti;
            }
          }
        }
      }
    }
  }

  // merge the two column-half lists of each row, emit 9 smallest (tie -> min idx)
#pragma unroll
  for (int s = 0; s < KNN; ++s) {
    mrgD[(wv * 16 + l16) * 18 + hi * KNN + s] = ld[s];
    mrgI[(wv * 16 + l16) * 18 + hi * KNN + s] = li[s];
  }
  __syncthreads();
  if (lane < 16) {
    const int base = (wv * 16 + lane) * 18;
    for (int s = 0; s < KNN; ++s) {
      float bd = FLT_BIG; int bi = 0x7fffffff; int bp = 0;
      for (int p = 0; p < 18; ++p) {
        float d = mrgD[base + p]; int j = mrgI[base + p];
        if (d < bd || (d == bd && j < bi)) { bd = d; bi = j; bp = p; }
      }
      mrgD[base + bp] = FLT_BIG;
      knn_idx[(size_t)(i0 + wv * 16 + lane) * KNN + s] = bi;
    }
  }
}

// ------------------------------------------- BN stats over y = P[n]+Q[idx]+bconv
__global__ __launch_bounds__(256) void bnstats_kernel(
    const float* __restrict__ P, const float* __restrict__ Q,
    const int* __restrict__ idx, const float* __restrict__ bconv,
    float* __restrict__ psum, float* __restrict__ psumsq) {
  int t = threadIdx.x;
  int n0 = blockIdx.x * 64;
  float s[3] = {0.f, 0.f, 0.f}, ss[3] = {0.f, 0.f, 0.f}, bc[3];
#pragma unroll
  for (int i = 0; i < 3; ++i) bc[i] = bconv[t + i * 256];
  for (int n = n0; n < n0 + 64; ++n) {
    float p[3];
#pragma unroll
    for (int i = 0; i < 3; ++i) p[i] = P[(size_t)n * CH2 + t + i * 256];
#pragma unroll
    for (int k = 0; k < KNN; ++k) {
      int j = idx[n * KNN + k];
#pragma unroll
      for (int i = 0; i < 3; ++i) {
        float y = p[i] + Q[(size_t)j * CH2 + t + i * 256] + bc[i];
        s[i] += y; ss[i] += y * y;
      }
    }
  }
#pragma unroll
  for (int i = 0; i < 3; ++i) {
    psum[blockIdx.x * CH2 + t + i * 256]   = s[i];
    psumsq[blockIdx.x * CH2 + t + i * 256] = ss[i];
  }
}

__global__ void bnfinalize_kernel(const float* __restrict__ psum,
                                  const float* __restrict__ psumsq,
                                  const float* __restrict__ gamma,
                                  const float* __restrict__ beta,
                                  const float* __restrict__ bconv,
                                  float* __restrict__ bn_a,
                                  float* __restrict__ bn_sh) {
  int c = blockIdx.x * blockDim.x + threadIdx.x;
  if (c < CH2) {
    float s = 0.f, ss = 0.f;
    for (int b = 0; b < NPTS / 64; ++b) {
      s += psum[b * CH2 + c];
      ss += psumsq[b * CH2 + c];
    }
    const float inv_cnt = 1.f / (float)((size_t)NPTS * KNN);
    float mean = s * inv_cnt;
    float var = ss * inv_cnt - mean * mean;
    float a = gamma[c] * rsqrtf(var + BN_EPS);
    bn_a[c] = a;
    bn_sh[c] = beta[c] + a * (bconv[c] - mean);  // y_bn = a*(P+Q) + bn_sh
  }
}

// --------------------------------------------- g[n,c] = max_k relu(BN(y[n,k,c]))
__global__ __launch_bounds__(256) void gmax_kernel(
    const float* __restrict__ P, const float* __restrict__ Q,
    const int* __restrict__ idx, const float* __restrict__ bn_a,
    const float* __restrict__ bn_sh, float* __restrict__ g) {
  int t = threadIdx.x;
  int n0 = blockIdx.x * 64;
  float a[3], sh[3];
#pragma unroll
  for (int i = 0; i < 3; ++i) { a[i] = bn_a[t + i * 256]; sh[i] = bn_sh[t + i * 256]; }
  for (int n = n0; n < n0 + 64; ++n) {
    float p[3], m[3];
#pragma unroll
    for (int i = 0; i < 3; ++i) {
      p[i] = P[(size_t)n * CH2 + t + i * 256];
      m[i] = -FLT_BIG;
    }
#pragma unroll
    for (int k = 0; k < KNN; ++k) {
      int j = idx[n * KNN + k];
#pragma unroll
      for (int i = 0; i < 3; ++i) {
        float y = fmaf(a[i], p[i] + Q[(size_t)j * CH2 + t + i * 256], sh[i]);
        m[i] = fmaxf(m[i], y);
      }
    }
#pragma unroll
    for (int i = 0; i < 3; ++i)
      g[(size_t)n * CH2 + t + i * 256] = fmaxf(m[i], 0.f);  // relu commutes with max
  }
}

// --------------------------------------------------------------------- launcher
extern "C" void kernel_launch(void* const* d_in, const int* in_sizes, int n_in,
                              void* d_out, int out_size, void* d_ws,
                              size_t ws_size, hipStream_t stream) {
  (void)in_sizes; (void)n_in; (void)out_size; (void)ws_size;
  const float* x     = (const float*)d_in[0];
  const float* w1    = (const float*)d_in[1];
  const float* b1    = (const float*)d_in[2];
  const float* wconv = (const float*)d_in[3];
  const float* bconv = (const float*)d_in[4];
  const float* gamma = (const float*)d_in[5];
  const float* beta  = (const float*)d_in[6];
  const float* w2    = (const float*)d_in[7];
  const float* b2    = (const float*)d_in[8];
  float* out = (float*)d_out;

  // workspace carve-up (floats); total ~51.5M floats (~206 MB)
  float* ws = (float*)d_ws;
  size_t off = 0;
  float* h    = ws + off; off += (size_t)NPTS * CH;        // fc1 output
  float* hnb  = ws + off; off += (size_t)NPTS * CH;        // column-normalized h
  float* P    = ws + off; off += (size_t)NPTS * CH2;       // h @ (Wa - Wb)
  float* Q    = ws + off; off += (size_t)NPTS * CH2;       // h @ Wb
  float* g    = ws + off; off += (size_t)NPTS * CH2;       // max-aggregated
  float* w1t  = ws + off; off += (size_t)CH * CH;
  float* wb1  = ws + off; off += (size_t)CH * CH2;
  float* wb2  = ws + off; off += (size_t)CH * CH2;
  float* w2t  = ws + off; off += (size_t)CH2 * CH;
  float* colpart = ws + off; off += (size_t)(NPTS / 64) * CH;
  float* invn    = ws + off; off += CH;
  float* sqrow   = ws + off; off += NPTS;
  float* psum    = ws + off; off += (size_t)(NPTS / 64) * CH2;
  float* psumsq  = ws + off; off += (size_t)(NPTS / 64) * CH2;
  float* bn_a    = ws + off; off += CH2;
  float* bn_sh   = ws + off; off += CH2;
  int* idx = (int*)(ws + off); off += (size_t)NPTS * KNN;

  // weight prep
  transpose_kernel<<<(CH * CH + 255) / 256, 256, 0, stream>>>(w1, w1t, CH, CH);
  split_wconv_kernel<<<(CH * CH2 + 255) / 256, 256, 0, stream>>>(wconv, wb1, wb2);
  transpose_kernel<<<(CH * CH2 + 255) / 256, 256, 0, stream>>>(w2, w2t, CH, CH2);

  // fc1: h = relu(x @ w1^T + b1)
  gemm_wmma_kernel<<<dim3(NPTS / 64, CH / 64), 128, 0, stream>>>(
      x, w1t, b1, nullptr, h, NPTS, CH, CH, 1);

  // column-L2 normalize h -> hn, and row sums sq
  colsumsq_kernel<<<NPTS / 64, 128, 0, stream>>>(h, colpart);
  colreduce_kernel<<<(CH + 127) / 128, 128, 0, stream>>>(colpart, invn);
  rownorm_kernel<<<NPTS, 128, 0, stream>>>(h, invn, hnb, sqrow);

  // EdgeConv linearization: P = h @ (Wa-Wb), Q = h @ Wb
  gemm_wmma_kernel<<<dim3(NPTS / 64, CH2 / 64), 128, 0, stream>>>(
      h, wb1, nullptr, nullptr, P, NPTS, CH, CH2, 0);
  gemm_wmma_kernel<<<dim3(NPTS / 64, CH2 / 64), 128, 0, stream>>>(
      h, wb2, nullptr, nullptr, Q, NPTS, CH, CH2, 0);

  // fused tiled KNN (dominant: ~206 GFLOP of fp32 WMMA)
  knn_kernel<<<NPTS / 64, 128, 0, stream>>>(hnb, sqrow, idx);

  // BN stats, finalize, max-over-k
  bnstats_kernel<<<NPTS / 64, 256, 0, stream>>>(P, Q, idx, bconv, psum, psumsq);
  bnfinalize_kernel<<<(CH2 + 255) / 256, 256, 0, stream>>>(
      psum, psumsq, gamma, beta, bconv, bn_a, bn_sh);
  gmax_kernel<<<NPTS / 64, 256, 0, stream>>>(P, Q, idx, bn_a, bn_sh, g);

  // fc2 + residual: out = relu(g @ w2^T + b2) + x
  gemm_wmma_kernel<<<dim3(NPTS / 64, CH / 64), 128, 0, stream>>>(
      g, w2t, b2, x, out, NPTS, CH2, CH, 1);
}